// FreqMaskPredictor_53712861003902
// MI455X (gfx1250) — compile-verified
//
#include <hip/hip_runtime.h>
#include <math.h>

typedef __attribute__((ext_vector_type(2))) float v2f;
typedef __attribute__((ext_vector_type(8))) float v8f;

#define BATCH 64
#define FEAT  512
#define NB    8
#define HDIM  1024
#define WHALF 513   // W/2 + 1
#define NPIX  (HDIM * WHALF)

// ---------------------------------------------------------------------------
// Kernel 1: band_weights = softmax(feat @ w^T + b)   -> d_ws[64*8] f32
// Computed transposed: D = Wpad(16x512) x feat^T(512x64) via f32 WMMA 16x16x4.
// Per the CDNA5 C/D layout, lane L (0..15) of wave `wv` then holds all 8 band
// logits of batch (wv*16 + L) in acc[0..7] -> in-lane softmax, no shuffles.
// ---------------------------------------------------------------------------
__global__ __launch_bounds__(128) void FreqMask_bw_kernel(
    const float* __restrict__ feat,   // [64,512]
    const float* __restrict__ wmat,   // [8,512]
    const float* __restrict__ bias,   // [8]
    float* __restrict__ bw)           // [64,8] out
{
  const int wave  = threadIdx.x >> 5;
  const int lane  = threadIdx.x & 31;
  const int half  = lane >> 4;      // 0: K+0/K+1, 1: K+2/K+3
  const int idx16 = lane & 15;

  v8f acc = {};

#if __has_builtin(__builtin_amdgcn_wmma_f32_16x16x4_f32)
  // A = wmat padded to 16 rows (rows 8..15 are zero).
  const int   arow  = (idx16 < NB) ? idx16 : 0;
  const float amask = (idx16 < NB) ? 1.0f : 0.0f;
  const float* __restrict__ arowp = wmat + arow * FEAT + half * 2;
  // B = feat^T : column n of B-tile = batch (wave*16 + n).
  const int batch = wave * 16 + idx16;
  const float* __restrict__ bcolp = feat + batch * FEAT + half * 2;

  for (int k = 0; k < FEAT; k += 4) {
    v2f a, b;
    a.x = arowp[k]     * amask;
    a.y = arowp[k + 1] * amask;
    b.x = bcolp[k];
    b.y = bcolp[k + 1];
    // 8 args: (neg_a, A, neg_b, B, c_mod, C, reuse_a, reuse_b)
    acc = __builtin_amdgcn_wmma_f32_16x16x4_f32(
        false, a, false, b, (short)0, acc, false, false);
  }
#else
  // Exact scalar fallback (same math, no matrix unit).
  if (lane < 16) {
    const int batch = wave * 16 + idx16;
    const float* __restrict__ fp = feat + batch * FEAT;
    for (int j = 0; j < NB; ++j) {
      float s = 0.0f;
      const float* __restrict__ wp = wmat + j * FEAT;
      for (int k = 0; k < FEAT; ++k) s = fmaf(fp[k], wp[k], s);
      acc[j] = s;
    }
  }
#endif

  if (lane < 16) {
    const int batch = wave * 16 + idx16;
    float l[NB];
    float m = -INFINITY;
#pragma unroll
    for (int j = 0; j < NB; ++j) { l[j] = acc[j] + bias[j]; m = fmaxf(m, l[j]); }
    float s = 0.0f;
#pragma unroll
    for (int j = 0; j < NB; ++j) { l[j] = expf(l[j] - m); s += l[j]; }
    const float inv = 1.0f / s;
#pragma unroll
    for (int j = 0; j < NB; ++j) bw[batch * NB + j] = l[j] * inv;
  }
}

// ---------------------------------------------------------------------------
// Kernel 2: streaming broadcast. Each thread owns one (h,w) pixel, computes
// its radial band ONCE, then writes band_weights[b][band] for all 64 batches
// with non-temporal stores (write-once 134.5 MB -> pure store bandwidth).
// ---------------------------------------------------------------------------
__global__ __launch_bounds__(256) void FreqMask_fill_kernel(
    const float* __restrict__ bw,    // [64,8]
    float* __restrict__ out)         // [64,1,1024,513]
{
  __shared__ float sbw[BATCH * NB];  // 2 KB
  const int tid = threadIdx.x;
  sbw[tid]       = bw[tid];
  sbw[tid + 256] = bw[tid + 256];
  __syncthreads();

  const int p = blockIdx.x * 256 + tid;
  if (p >= NPIX) return;
  const int h  = p / WHALF;
  const int wi = p - h * WHALF;

  // rfftfreq / fftfreq values are exact (k / 2^10).
  const float u  = (float)wi * (1.0f / 1024.0f);
  const float vf = (h < HDIM / 2) ? (float)h * (1.0f / 1024.0f)
                                  : (float)(h - HDIM) * (1.0f / 1024.0f);
  const float r    = sqrtf(u * u + vf * vf);
  const float maxr = sqrtf(0.5f);    // radius.max() (exact same f32 value)

  // band = #{ i in 1..7 : r >= maxr*(i/8) }; same boundary expr as reference.
  int band = 0;
#pragma unroll
  for (int i = 1; i < NB; ++i)
    band += (r >= maxr * ((float)i * (1.0f / NB))) ? 1 : 0;
  const bool in_band = (r < maxr);   // corner pixel r==maxr -> 0

  const size_t off = (size_t)h * WHALF + wi;
#pragma unroll 8
  for (int b = 0; b < BATCH; ++b) {
    const float val = in_band ? sbw[b * NB + band] : 0.0f;
    __builtin_nontemporal_store(val, out + (size_t)b * NPIX + off);
  }
}

extern "C" void kernel_launch(void* const* d_in, const int* in_sizes, int n_in,
                              void* d_out, int out_size, void* d_ws, size_t ws_size,
                              hipStream_t stream) {
  const float* feat = (const float*)d_in[0];  // [64,512]
  const float* wmat = (const float*)d_in[1];  // [8,512]
  const float* bias = (const float*)d_in[2];  // [8]
  // d_in[3]=H, d_in[4]=W are fixed scalars (1024,1024) -> compile-time consts.

  float* bw  = (float*)d_ws;        // 2 KB scratch for band weights
  float* out = (float*)d_out;       // [64,1,1024,513] f32

  FreqMask_bw_kernel<<<1, 128, 0, stream>>>(feat, wmat, bias, bw);

  const int nblocks = (NPIX + 255) / 256;   // 2052
  FreqMask_fill_kernel<<<nblocks, 256, 0, stream>>>(bw, out);
}